// DualHeadGATModel_18880676233460
// MI455X (gfx1250) — compile-verified
//
#include <hip/hip_runtime.h>

// 5-layer GAT forward (PyG GATConv semantics) for MI455X / gfx1250.
// GEMMs for layers 1-3 use V_WMMA_F32_16X16X4_F32 (full fp32 matrix cores).
// Edge phases are memory-bound scatter/gather with float atomics.

#define NNODES 20000
#define NEDGES 320000
#define ETOT   (NEDGES + NNODES)

typedef __attribute__((ext_vector_type(2))) float v2f;
typedef __attribute__((ext_vector_type(8))) float v8f;

// ---------------- Layer 0 GEMM: g[N,512] = x[N,2] @ W0[2,512] ----------------
__global__ void gemm_cin2(const float* __restrict__ x, const float* __restrict__ W,
                          float* __restrict__ g) {
    int tid = blockIdx.x * blockDim.x + threadIdx.x;
    if (tid >= NNODES * 512) return;
    int j = tid & 511;
    int i = tid >> 9;
    g[tid] = x[i * 2] * W[j] + x[i * 2 + 1] * W[512 + j];
}

// ---------------- Layer 4 GEMM: g[N,2] = h[N,512] @ W4[512,2] ----------------
__global__ void gemm_cout2(const float* __restrict__ h, const float* __restrict__ W,
                           float* __restrict__ g) {
    int i = blockIdx.x * blockDim.x + threadIdx.x;
    if (i >= NNODES) return;
    const float* hp = h + (size_t)i * 512;
    float a0 = 0.f, a1 = 0.f;
    #pragma unroll 8
    for (int k = 0; k < 512; k++) {
        float v = hp[k];
        a0 += v * W[2 * k];
        a1 += v * W[2 * k + 1];
    }
    g[2 * i]     = a0;
    g[2 * i + 1] = a1;
}

// -------- fp32 WMMA GEMM: C[M,N] = A[M,K] @ B[K,N]; M%16==0, K%4==0, N%64==0 --------
// One wave computes a 16x64 tile with 4 interleaved accumulators.
__global__ __launch_bounds__(256) void gemm_wmma_f32(const float* __restrict__ A,
                                                     const float* __restrict__ B,
                                                     float* __restrict__ C,
                                                     int M, int K, int N) {
    const int lane   = threadIdx.x & 31;
    const int wave   = (blockIdx.x * blockDim.x + threadIdx.x) >> 5;
    const int tilesN = N >> 6;
    const int tm = wave / tilesN;
    const int tn = wave - tm * tilesN;
    if (tm * 16 >= M) return;
    const int m0   = tm << 4;
    const int n0   = tn << 6;
    const int half = lane >> 4;   // 0: K+0/1, 1: K+2/3 (ISA 16x4 f32 A layout)
    const int l15  = lane & 15;

    const float* Arow = A + (size_t)(m0 + l15) * K + half * 2;
    const float* Bcol = B + n0 + l15;

    v8f acc0 = {}, acc1 = {}, acc2 = {}, acc3 = {};
    for (int k = 0; k < K; k += 4) {
        v2f a;
        a.x = Arow[k];
        a.y = Arow[k + 1];
        const float* B0 = Bcol + (size_t)(k + half * 2) * N;
        const float* B1 = B0 + N;
        v2f b0, b1, b2, b3;
        b0.x = B0[0];  b0.y = B1[0];
        b1.x = B0[16]; b1.y = B1[16];
        b2.x = B0[32]; b2.y = B1[32];
        b3.x = B0[48]; b3.y = B1[48];
        acc0 = __builtin_amdgcn_wmma_f32_16x16x4_f32(false, a, false, b0, (short)0, acc0, false, false);
        acc1 = __builtin_amdgcn_wmma_f32_16x16x4_f32(false, a, false, b1, (short)0, acc1, false, false);
        acc2 = __builtin_amdgcn_wmma_f32_16x16x4_f32(false, a, false, b2, (short)0, acc2, false, false);
        acc3 = __builtin_amdgcn_wmma_f32_16x16x4_f32(false, a, false, b3, (short)0, acc3, false, false);
    }
    // C/D layout: VGPR i holds rows m0+i (lanes 0-15) and m0+8+i (lanes 16-31)
    float* Crow = C + n0 + l15;
    const int mbase = m0 + half * 8;
    #pragma unroll
    for (int i = 0; i < 8; i++) {
        float* Ci = Crow + (size_t)(mbase + i) * N;
        Ci[0]  = acc0[i];
        Ci[16] = acc1[i];
        Ci[32] = acc2[i];
        Ci[48] = acc3[i];
    }
}

// ---------------- attention logits: e_src/e_dst[n,h] = sum_c g[n,h,c]*a[h,c] ----------------
__global__ void attn_logits(const float* __restrict__ g, const float* __restrict__ as_,
                            const float* __restrict__ ad_, float* __restrict__ esrc,
                            float* __restrict__ edst, int logH, int C) {
    int tid = blockIdx.x * blockDim.x + threadIdx.x;
    int H = 1 << logH;
    if (tid >= NNODES * H) return;
    int h = tid & (H - 1);
    int i = tid >> logH;
    const float* gp  = g + (size_t)(i * H + h) * C;
    const float* asp = as_ + h * C;
    const float* adp = ad_ + h * C;
    float ss = 0.f, sd = 0.f;
    for (int c = 0; c < C; c++) {
        float v = gp[c];
        ss += v * asp[c];
        sd += v * adp[c];
    }
    esrc[tid] = ss;
    edst[tid] = sd;
}

__global__ void init_softmax(float* __restrict__ amax, float* __restrict__ denom, int n) {
    int tid = blockIdx.x * blockDim.x + threadIdx.x;
    if (tid >= n) return;
    ((int*)amax)[tid] = 0xFF800000;  // -inf
    denom[tid] = 0.f;
}

__device__ __forceinline__ void edge_nodes(int e, const int* __restrict__ ei, int& s, int& d) {
    if (e < NEDGES) { s = ei[e]; d = ei[NEDGES + e]; }
    else            { s = e - NEDGES; d = s; }       // self loops
}

__device__ __forceinline__ float leaky02(float t) { return t > 0.f ? t : 0.2f * t; }

// segment max via float-as-int atomic trick (-> global_atomic_max_i32 / min_u32)
__global__ void edge_max(const int* __restrict__ ei, const float* __restrict__ esrc,
                         const float* __restrict__ edst, float* __restrict__ amax, int logH) {
    int tid = blockIdx.x * blockDim.x + threadIdx.x;
    int H = 1 << logH;
    if (tid >= (ETOT << logH)) return;
    int h = tid & (H - 1);
    int e = tid >> logH;
    int s, d;
    edge_nodes(e, ei, s, d);
    float al = leaky02(esrc[(s << logH) + h] + edst[(d << logH) + h]);
    int* addr = (int*)&amax[(d << logH) + h];
    if (al >= 0.f) atomicMax(addr, __float_as_int(al));
    else           atomicMin((unsigned int*)addr, __float_as_uint(al));
}

__global__ void edge_exp(const int* __restrict__ ei, const float* __restrict__ esrc,
                         const float* __restrict__ edst, const float* __restrict__ amax,
                         float* __restrict__ denom, float* __restrict__ ex, int logH) {
    int tid = blockIdx.x * blockDim.x + threadIdx.x;
    int H = 1 << logH;
    if (tid >= (ETOT << logH)) return;
    int h = tid & (H - 1);
    int e = tid >> logH;
    int s, d;
    edge_nodes(e, ei, s, d);
    int dh = (d << logH) + h;
    float al = leaky02(esrc[(s << logH) + h] + edst[dh]);
    float exv = expf(al - amax[dh]);
    ex[tid] = exv;
    atomicAdd(&denom[dh], exv);
}

__global__ void zero_f32(float* __restrict__ p, int n) {
    int tid = blockIdx.x * blockDim.x + threadIdx.x;
    if (tid < n) p[tid] = 0.f;
}

// message scatter: out[dst, h, c] += g[src, h, c] * ex/(denom+eps)
__global__ void scatter_msg(const int* __restrict__ ei, const float* __restrict__ g,
                            const float* __restrict__ ex, const float* __restrict__ denom,
                            float* __restrict__ out, int logHC, int logC) {
    int tid = blockIdx.x * blockDim.x + threadIdx.x;
    if (tid >= (int)(ETOT << logHC)) return;
    int idx = tid & ((1 << logHC) - 1);
    int e   = tid >> logHC;
    int hh  = idx >> logC;
    int logH = logHC - logC;
    int s, d;
    edge_nodes(e, ei, s, d);
    float coef = ex[(e << logH) + hh] / (denom[(d << logH) + hh] + 1e-16f);
    atomicAdd(&out[(d << logHC) + idx], g[(s << logHC) + idx] * coef);
}

__global__ void bias_relu(float* __restrict__ h, const float* __restrict__ b, int logD) {
    int tid = blockIdx.x * blockDim.x + threadIdx.x;
    if (tid >= (NNODES << logD)) return;
    float v = h[tid] + b[tid & ((1 << logD) - 1)];
    h[tid] = v > 0.f ? v : 0.f;
}

// ---------------- host-side edge-phase driver ----------------
static void run_edge_phase(const int* ei, const float* g, const float* as_, const float* ad_,
                           const float* b, float* out, float* esrc, float* edst, float* amax,
                           float* denom, float* ex, int logH, int C, int logC,
                           hipStream_t stream) {
    const int logHC = logH + logC;
    const int nh  = NNODES << logH;
    const int eth = ETOT << logH;
    const int outn = NNODES << logHC;
    const long tot = (long)ETOT << logHC;

    attn_logits<<<(nh + 255) / 256, 256, 0, stream>>>(g, as_, ad_, esrc, edst, logH, C);
    init_softmax<<<(nh + 255) / 256, 256, 0, stream>>>(amax, denom, nh);
    edge_max<<<(eth + 255) / 256, 256, 0, stream>>>(ei, esrc, edst, amax, logH);
    edge_exp<<<(eth + 255) / 256, 256, 0, stream>>>(ei, esrc, edst, amax, denom, ex, logH);
    zero_f32<<<(outn + 255) / 256, 256, 0, stream>>>(out, outn);
    scatter_msg<<<(unsigned)((tot + 255) / 256), 256, 0, stream>>>(ei, g, ex, denom, out, logHC, logC);
    bias_relu<<<(outn + 255) / 256, 256, 0, stream>>>(out, b, logHC);
}

extern "C" void kernel_launch(void* const* d_in, const int* in_sizes, int n_in,
                              void* d_out, int out_size, void* d_ws, size_t ws_size,
                              hipStream_t stream) {
    const float* x  = (const float*)d_in[0];
    const int*   ei = (const int*)d_in[1];
    // d_in[2] = fixed_tof_mask (unused by reference math)
    const float *W[5], *AS[5], *AD[5], *Bb[5];
    for (int i = 0; i < 5; i++) {
        W[i]  = (const float*)d_in[3 + 4 * i];
        AS[i] = (const float*)d_in[4 + 4 * i];
        AD[i] = (const float*)d_in[5 + 4 * i];
        Bb[i] = (const float*)d_in[6 + 4 * i];
    }

    float* ws    = (float*)d_ws;
    float* g     = ws;                               // N*512
    float* hA    = g    + (size_t)NNODES * 512;      // N*512
    float* hB    = hA   + (size_t)NNODES * 512;      // N*512
    float* esrc  = hB   + (size_t)NNODES * 512;      // N*8
    float* edst  = esrc + (size_t)NNODES * 8;        // N*8
    float* amax  = edst + (size_t)NNODES * 8;        // N*8
    float* denom = amax + (size_t)NNODES * 8;        // N*8
    float* ex    = denom + (size_t)NNODES * 8;       // ETOT*8

    // ---- layer 0: (2 -> 64x8 concat) ----
    gemm_cin2<<<(NNODES * 512 + 255) / 256, 256, 0, stream>>>(x, W[0], g);
    run_edge_phase(ei, g, AS[0], AD[0], Bb[0], hA, esrc, edst, amax, denom, ex,
                   /*logH=*/3, /*C=*/64, /*logC=*/6, stream);

    // ---- layers 1..3: (512 -> 64x8 concat), fp32 WMMA GEMM ----
    float* cur = hA;
    float* nxt = hB;
    for (int L = 1; L <= 3; L++) {
        // waves = (20000/16)*(512/64) = 10000 -> 1250 blocks of 8 waves
        gemm_wmma_f32<<<1250, 256, 0, stream>>>(cur, W[L], g, NNODES, 512, 512);
        run_edge_phase(ei, g, AS[L], AD[L], Bb[L], nxt, esrc, edst, amax, denom, ex,
                       3, 64, 6, stream);
        float* t = cur; cur = nxt; nxt = t;
    }

    // ---- layer 4: (512 -> 2, 1 head, mean==identity), output straight to d_out ----
    gemm_cout2<<<(NNODES + 255) / 256, 256, 0, stream>>>(cur, W[4], g);
    run_edge_phase(ei, g, AS[4], AD[4], Bb[4], (float*)d_out, esrc, edst, amax, denom, ex,
                   /*logH=*/0, /*C=*/2, /*logC=*/1, stream);
}